// GINEncoder_34385508172366
// MI455X (gfx1250) — compile-verified
//
#include <hip/hip_runtime.h>

#define NN      100000
#define NE      1600000
#define DD      128
#define LDSW    144          // 128+16: half-wave row offset 2*144 mod 64 banks = 32 -> no conflict
#define SBLK    512          // stats partial blocks

typedef float v2f __attribute__((ext_vector_type(2)));
typedef float v8f __attribute__((ext_vector_type(8)));

// ---------------- utility: zero a float buffer (float4 granularity) ----------------
__global__ __launch_bounds__(256) void zero_kernel(float4* __restrict__ p, int n4) {
    int i = blockIdx.x * 256 + threadIdx.x;
    if (i < n4) p[i] = make_float4(0.f, 0.f, 0.f, 0.f);
}

// ---------------- edge scatter-add: agg[dst] += h[src] ----------------
// 32 threads per edge, each moves one float4 chunk; f32 HW atomics hit L2 (agg fits in 192MB L2).
__global__ __launch_bounds__(256) void scatter_kernel(const float* __restrict__ h,
                                                      const int* __restrict__ ei,
                                                      float* __restrict__ agg) {
    int t = blockIdx.x * 256 + threadIdx.x;
    int e = t >> 5;
    if (e >= NE) return;
    int chunk = (t & 31) * 4;
    int src = ei[e];
    int dst = ei[NE + e];
    const float4 v = *(const float4*)(h + (size_t)src * DD + chunk);
    float* p = agg + (size_t)dst * DD + chunk;
    unsafeAtomicAdd(p + 0, v.x);
    unsafeAtomicAdd(p + 1, v.y);
    unsafeAtomicAdd(p + 2, v.z);
    unsafeAtomicAdd(p + 3, v.w);
}

// ---------------- fused GEMM: out = f(A) @ W + bias ----------------
// MODE 0: f(A) = A + Agg                 (GIN self + aggregate)
// MODE 1: f(A) = relu(A*scale + shift)   (BN affine + ReLU fused into A load)
// Block = 256 threads = 8 waves; block computes 128 rows x 128 cols; W staged in LDS.
// Per-wave 16x128 strip via V_WMMA_F32_16X16X4_F32, K-loop of 32 per 16-wide N tile.
template <int MODE>
__global__ __launch_bounds__(256) void gemm_kernel(const float* __restrict__ A,
                                                   const float* __restrict__ Agg,
                                                   const float* __restrict__ scale,
                                                   const float* __restrict__ shift,
                                                   const float* __restrict__ W,
                                                   const float* __restrict__ bias,
                                                   float* __restrict__ out) {
    __shared__ float sW[DD * LDSW];
    __shared__ float sBias[DD];

    int tid = threadIdx.x;
    for (int idx = tid; idx < DD * DD; idx += 256) {
        int k = idx >> 7, n = idx & 127;
        sW[k * LDSW + n] = W[idx];
    }
    if (tid < DD) sBias[tid] = bias[tid];
    __syncthreads();

    const int wave = tid >> 5;
    const int lane = tid & 31;
    const int hi   = lane >> 4;   // half-wave: 0 or 1
    const int lo   = lane & 15;
    const int m0   = blockIdx.x * 128 + wave * 16;
    int row  = m0 + lo;
    int rowc = row < NN ? row : NN - 1;   // clamp loads; stores guarded below

    // Preload this wave's 16x128 A strip as WMMA A-fragments (ISA 16x4 f32 layout).
    v2f afrag[32];
    const float* arow = A + (size_t)rowc * DD;
    const float* grow = (MODE == 0) ? (Agg + (size_t)rowc * DD) : nullptr;
#pragma unroll
    for (int kb = 0; kb < 32; ++kb) {
        int c = 4 * kb + 2 * hi;
        float x0 = arow[c], x1 = arow[c + 1];
        if (MODE == 0) {
            x0 += grow[c];
            x1 += grow[c + 1];
        } else {
            x0 = fmaxf(x0 * scale[c] + shift[c], 0.f);
            x1 = fmaxf(x1 * scale[c + 1] + shift[c + 1], 0.f);
        }
        afrag[kb] = (v2f){x0, x1};
    }

    for (int nt = 0; nt < 8; ++nt) {
        const int n = nt * 16 + lo;
        const float bv = sBias[n];
        v8f acc = {bv, bv, bv, bv, bv, bv, bv, bv};
#pragma unroll
        for (int kb = 0; kb < 32; ++kb) {
            int k0 = 4 * kb + 2 * hi;
            v2f b;
            b.x = sW[k0 * LDSW + n];
            b.y = sW[(k0 + 1) * LDSW + n];
            acc = __builtin_amdgcn_wmma_f32_16x16x4_f32(
                false, afrag[kb], false, b, (short)0, acc, false, false);
        }
#pragma unroll
        for (int r = 0; r < 8; ++r) {
            int m = m0 + r + 8 * hi;     // C/D layout: VGPR r -> M=r (lo half) / r+8 (hi half)
            if (m < NN) out[(size_t)m * DD + n] = acc[r];
        }
    }
}

// ---------------- BN stats, stage 1: deterministic per-block partial sums ----------------
__global__ __launch_bounds__(128) void stats_partial_kernel(const float* __restrict__ t,
                                                            float* __restrict__ ps,
                                                            float* __restrict__ pss) {
    int col = threadIdx.x;
    float s = 0.f, ss = 0.f;
    for (int r = blockIdx.x; r < NN; r += gridDim.x) {
        float v = t[(size_t)r * DD + col];
        s += v;
        ss += v * v;
    }
    ps[blockIdx.x * DD + col]  = s;
    pss[blockIdx.x * DD + col] = ss;
}

// ---------------- BN stats, stage 2: finalize, fold gamma/beta into scale/shift ----------------
__global__ __launch_bounds__(128) void stats_finalize_kernel(const float* __restrict__ ps,
                                                             const float* __restrict__ pss,
                                                             const float* __restrict__ gamma,
                                                             const float* __restrict__ beta,
                                                             float* __restrict__ scale,
                                                             float* __restrict__ shift) {
    int col = threadIdx.x;
    float s = 0.f, ss = 0.f;
    for (int b = 0; b < SBLK; ++b) {
        s += ps[b * DD + col];
        ss += pss[b * DD + col];
    }
    const float inv = 1.f / (float)NN;
    float mu   = s * inv;
    float var  = ss * inv - mu * mu;
    float rstd = rsqrtf(var + 1e-5f);
    float sc   = gamma[col] * rstd;
    scale[col] = sc;
    shift[col] = beta[col] - mu * sc;
}

// ---------------- in-place affine + ReLU (inter-layer BN) ----------------
__global__ __launch_bounds__(256) void apply_kernel(float* __restrict__ h,
                                                    const float* __restrict__ scale,
                                                    const float* __restrict__ shift) {
    int i = blockIdx.x * 256 + threadIdx.x;           // float4 index, NN*32 total
    if (i >= NN * (DD / 4)) return;
    int c = (i & 31) * 4;
    float4 v = *(float4*)(h + (size_t)i * 4);
    v.x = fmaxf(v.x * scale[c + 0] + shift[c + 0], 0.f);
    v.y = fmaxf(v.y * scale[c + 1] + shift[c + 1], 0.f);
    v.z = fmaxf(v.z * scale[c + 2] + shift[c + 2], 0.f);
    v.w = fmaxf(v.w * scale[c + 3] + shift[c + 3], 0.f);
    *(float4*)(h + (size_t)i * 4) = v;
}

// ---------------- final row-wise L2 normalize (one wave32 per row) ----------------
__global__ __launch_bounds__(256) void normalize_kernel(const float* __restrict__ h,
                                                        float* __restrict__ out) {
    int wave = threadIdx.x >> 5;
    int lane = threadIdx.x & 31;
    int row = blockIdx.x * 8 + wave;
    if (row >= NN) return;
    const float4 v = *(const float4*)(h + (size_t)row * DD + lane * 4);
    float ss = v.x * v.x + v.y * v.y + v.z * v.z + v.w * v.w;
#pragma unroll
    for (int off = 16; off; off >>= 1) ss += __shfl_xor(ss, off, 32);
    float inv = 1.f / fmaxf(sqrtf(ss), 1e-12f);
    float4 o = {v.x * inv, v.y * inv, v.z * inv, v.w * inv};
    *(float4*)(out + (size_t)row * DD + lane * 4) = o;
}

extern "C" void kernel_launch(void* const* d_in, const int* in_sizes, int n_in,
                              void* d_out, int out_size, void* d_ws, size_t ws_size,
                              hipStream_t stream) {
    const float* x   = (const float*)d_in[0];
    const int*   ei  = (const int*)d_in[1];
    const float* W1  = (const float*)d_in[2];
    const float* b1  = (const float*)d_in[3];
    const float* g1  = (const float*)d_in[4];
    const float* be1 = (const float*)d_in[5];
    const float* W2  = (const float*)d_in[6];
    const float* b2  = (const float*)d_in[7];
    const float* go  = (const float*)d_in[8];
    const float* bo  = (const float*)d_in[9];
    float* out = (float*)d_out;
    float* ws  = (float*)d_ws;

    const size_t HBUF = (size_t)NN * DD;              // 12.8M floats
    float* bufA  = ws;
    float* bufB  = ws + HBUF;
    float* agg   = ws + 2 * HBUF;
    float* ps    = ws + 3 * HBUF;
    float* pss   = ps + SBLK * DD;
    float* scale = pss + SBLK * DD;
    float* shift = scale + DD;
    const size_t need = (size_t)(shift + DD - ws) * sizeof(float);
    if (ws_size < need) return;                        // scratch too small: bail

    float* t1 = out;                                   // reuse d_out as t1 scratch
    const int gemm_blocks = (NN + 127) / 128;
    const int zero4 = (int)(HBUF / 4);

    const float* hin = x;
    float* hbufs[3] = {bufA, bufB, bufA};

    for (int i = 0; i < 3; ++i) {
        float* hout = hbufs[i];
        // 1) agg = 0 ; agg += h[src] scattered to dst
        zero_kernel<<<(zero4 + 255) / 256, 256, 0, stream>>>((float4*)agg, zero4);
        scatter_kernel<<<(NE * 32) / 256, 256, 0, stream>>>(hin, ei, agg);
        // 2) t1 = (h + agg) @ W1[i] + b1[i]
        gemm_kernel<0><<<gemm_blocks, 256, 0, stream>>>(hin, agg, nullptr, nullptr,
                                                        W1 + (size_t)i * DD * DD,
                                                        b1 + (size_t)i * DD, t1);
        // 3) BN stats over t1 -> scale/shift (gamma=g1, beta=be1)
        stats_partial_kernel<<<SBLK, 128, 0, stream>>>(t1, ps, pss);
        stats_finalize_kernel<<<1, 128, 0, stream>>>(ps, pss, g1 + (size_t)i * DD,
                                                     be1 + (size_t)i * DD, scale, shift);
        // 4) hout = relu(bn(t1)) @ W2[i] + b2[i]
        gemm_kernel<1><<<gemm_blocks, 256, 0, stream>>>(t1, nullptr, scale, shift,
                                                        W2 + (size_t)i * DD * DD,
                                                        b2 + (size_t)i * DD, hout);
        // 5) inter-layer BN + ReLU applied in place
        if (i < 2) {
            stats_partial_kernel<<<SBLK, 128, 0, stream>>>(hout, ps, pss);
            stats_finalize_kernel<<<1, 128, 0, stream>>>(ps, pss, go + (size_t)i * DD,
                                                         bo + (size_t)i * DD, scale, shift);
            apply_kernel<<<(NN * (DD / 4) + 255) / 256, 256, 0, stream>>>(hout, scale, shift);
        }
        hin = hout;
    }
    // 6) row-wise L2 normalize -> d_out
    normalize_kernel<<<(NN + 7) / 8, 256, 0, stream>>>(hbufs[2], out);
}